// PathwiseFunction_38835094290932
// MI455X (gfx1250) — compile-verified
//
#include <hip/hip_runtime.h>

typedef __attribute__((ext_vector_type(2))) float v2f;
typedef __attribute__((ext_vector_type(8))) float v8f;

#define PW_D 8
#define PW_S 8
#define PW_F 1024
#define PW_R 2048

// Fused pathwise GP sampler:
//   out[s,n] = sum_f fw[s,f]*scale*cos((X/L)@omega[s] + phase[s,f])
//            + sum_r cw[s,r]*var*exp(-0.5*|X/L - Z/L|^2)
// One block = 16 rows of N; 8 waves; WMMA f32 16x16x4 for both GEMMs.
__global__ __launch_bounds__(256) void pathwise_fused(
    const float* __restrict__ X,      // [N, 8]
    const float* __restrict__ Z,      // [R, 8]
    const float* __restrict__ omega,  // [S, 8, F]
    const float* __restrict__ phase,  // [S, F]
    const float* __restrict__ fw,     // [S, F]
    const float* __restrict__ cw,     // [S, R]
    const float* __restrict__ var_p,  // scalar
    const float* __restrict__ len_p,  // scalar
    float* __restrict__ out,          // [S, N]
    int N)
{
  __shared__ float ldsX[16 * PW_D];       // X tile scaled by 1/L
  __shared__ float x2s[16];               // |x/L|^2 per row
  __shared__ float updAll[PW_S * 16];     // correction accum [s][row]

  const int tid  = threadIdx.x;
  const int wave = tid >> 5;              // 8 waves
  const int lane = tid & 31;
  const int m    = lane & 15;             // column index within WMMA tile
  const int hi   = lane >> 4;             // lane-half selector
  const int n0   = blockIdx.x * 16;

  const float variance = var_p[0];
  const float invL     = 1.0f / len_p[0];
  const float scale    = __fsqrt_rn(2.0f * variance / (float)PW_F);

  // ---- stage scaled X tile + zero the shared correction accumulator ----
  if (tid < 16 * PW_D) {
    const int row = tid >> 3, col = tid & 7;
    ldsX[tid] = X[(size_t)(n0 + row) * PW_D + col] * invL;
  }
  if (tid < PW_S * 16) updAll[tid] = 0.0f;
  __syncthreads();

  // ---- A operand: 16x4 f32 layout. lane holds row M=lane%16;
  //      VGPR0 = K (hi?2:0), VGPR1 = K (hi?3:1). Two chunks cover D=8. ----
  v2f a_lo, a_hi;
  a_lo.x = ldsX[m * 8 + (hi ? 2 : 0)];
  a_lo.y = ldsX[m * 8 + (hi ? 3 : 1)];
  a_hi.x = ldsX[m * 8 + (hi ? 6 : 4)];
  a_hi.y = ldsX[m * 8 + (hi ? 7 : 5)];

  // |x/L|^2 : lane l has half the d's, lane l^16 has the other half
  {
    float xp = a_lo.x * a_lo.x + a_lo.y * a_lo.y +
               a_hi.x * a_hi.x + a_hi.y * a_hi.y;
    float x2m = xp + __shfl_xor(xp, 16, 32);
    if (wave == 0 && hi == 0) x2s[m] = x2m;
  }
  __syncthreads();

  float x2r[8];
#pragma unroll
  for (int v = 0; v < 8; ++v) x2r[v] = x2s[v + (hi ? 8 : 0)];

  // ================= Phase 1: RFF prior (wave w owns s = w) =================
  const int s = wave;
  const float* __restrict__ omega_s = omega + (size_t)s * PW_D * PW_F;
  const float* __restrict__ phase_s = phase + (size_t)s * PW_F;
  const float* __restrict__ fw_s    = fw    + (size_t)s * PW_F;

  float acc[8];
#pragma unroll
  for (int v = 0; v < 8; ++v) acc[v] = 0.0f;

  for (int f0 = 0; f0 < PW_F; f0 += 16) {
    const int c = f0 + m;
    // B operand: 4x16. lane holds col N=lane%16; lanes<16: K=0,1; lanes>=16: K=2,3
    v2f b_lo, b_hi;
    b_lo.x = omega_s[(hi ? 2 : 0) * PW_F + c];
    b_lo.y = omega_s[(hi ? 3 : 1) * PW_F + c];
    b_hi.x = omega_s[(hi ? 6 : 4) * PW_F + c];
    b_hi.y = omega_s[(hi ? 7 : 5) * PW_F + c];

    v8f d = {0.f, 0.f, 0.f, 0.f, 0.f, 0.f, 0.f, 0.f};
    d = __builtin_amdgcn_wmma_f32_16x16x4_f32(false, a_lo, false, b_lo,
                                              (short)0, d, false, false);
    d = __builtin_amdgcn_wmma_f32_16x16x4_f32(false, a_hi, false, b_hi,
                                              (short)0, d, false, false);
    const float ph = phase_s[c];
    const float w  = fw_s[c];
#pragma unroll
    for (int v = 0; v < 8; ++v)
      acc[v] += w * __cosf(d[v] + ph);      // v_cos_f32 path
  }

  // reduce prior over the 16 f-columns held by each lane half
  float accR[8];
#pragma unroll
  for (int v = 0; v < 8; ++v) {
    float t = acc[v];
    t += __shfl_xor(t, 1, 32);
    t += __shfl_xor(t, 2, 32);
    t += __shfl_xor(t, 4, 32);
    t += __shfl_xor(t, 8, 32);
    accR[v] = t;
  }

  // ========== Phase 2: RBF correction (waves split R; all s kept) ==========
  float upd[PW_S][8];
#pragma unroll
  for (int s2 = 0; s2 < PW_S; ++s2)
#pragma unroll
    for (int v = 0; v < 8; ++v) upd[s2][v] = 0.0f;

  for (int t = wave; t < PW_R / 16; t += PW_S) {
    const int r = (t << 4) + m;
    const float* __restrict__ zr = Z + (size_t)r * PW_D;
    v2f b_lo, b_hi;
    b_lo.x = zr[hi ? 2 : 0] * invL;
    b_lo.y = zr[hi ? 3 : 1] * invL;
    b_hi.x = zr[hi ? 6 : 4] * invL;
    b_hi.y = zr[hi ? 7 : 5] * invL;

    float zp = b_lo.x * b_lo.x + b_lo.y * b_lo.y +
               b_hi.x * b_hi.x + b_hi.y * b_hi.y;
    const float z2 = zp + __shfl_xor(zp, 16, 32);

    v8f g = {0.f, 0.f, 0.f, 0.f, 0.f, 0.f, 0.f, 0.f};
    g = __builtin_amdgcn_wmma_f32_16x16x4_f32(false, a_lo, false, b_lo,
                                              (short)0, g, false, false);
    g = __builtin_amdgcn_wmma_f32_16x16x4_f32(false, a_hi, false, b_hi,
                                              (short)0, g, false, false);

    float kv[8];
#pragma unroll
    for (int v = 0; v < 8; ++v) {
      float sq = x2r[v] + z2 - 2.0f * g[v];
      sq = fmaxf(sq, 0.0f);
      kv[v] = variance * __expf(-0.5f * sq);  // v_exp_f32 path
    }
#pragma unroll
    for (int s2 = 0; s2 < PW_S; ++s2) {
      const float w = cw[(size_t)s2 * PW_R + r];
#pragma unroll
      for (int v = 0; v < 8; ++v) upd[s2][v] += w * kv[v];
    }
  }

  // reduce over the 16 r-columns per half, then combine waves via LDS atomics
#pragma unroll
  for (int s2 = 0; s2 < PW_S; ++s2) {
#pragma unroll
    for (int v = 0; v < 8; ++v) {
      float t = upd[s2][v];
      t += __shfl_xor(t, 1, 32);
      t += __shfl_xor(t, 2, 32);
      t += __shfl_xor(t, 4, 32);
      t += __shfl_xor(t, 8, 32);
      if (m == s2) atomicAdd(&updAll[s2 * 16 + v + 8 * hi], t);
    }
  }
  __syncthreads();

  // ---- final write: wave s writes its 16 rows (lane 0 -> rows 0..7, lane 16 -> 8..15)
  if (m == 0) {
#pragma unroll
    for (int v = 0; v < 8; ++v) {
      const int row = v + 8 * hi;
      out[(size_t)s * N + n0 + row] = scale * accR[v] + updAll[s * 16 + row];
    }
  }
}

extern "C" void kernel_launch(void* const* d_in, const int* in_sizes, int n_in,
                              void* d_out, int out_size, void* d_ws, size_t ws_size,
                              hipStream_t stream) {
  const float* X     = (const float*)d_in[0];  // [N,8]
  const float* Z     = (const float*)d_in[1];  // [R,8]
  const float* omega = (const float*)d_in[2];  // [S,8,F]
  const float* phase = (const float*)d_in[3];  // [S,F]
  const float* fw    = (const float*)d_in[4];  // [S,F]
  const float* cw    = (const float*)d_in[5];  // [S,R]
  const float* var_p = (const float*)d_in[6];
  const float* len_p = (const float*)d_in[7];
  float* out = (float*)d_out;

  const int N = in_sizes[0] / PW_D;            // 16384
  dim3 grid(N / 16), block(256);
  pathwise_fused<<<grid, block, 0, stream>>>(X, Z, omega, phase, fw, cw,
                                             var_p, len_p, out, N);
}